// JKNet_5634997092461
// MI455X (gfx1250) — compile-verified
//
#include <hip/hip_runtime.h>

// JKNet (GraphConv x7 + jumping-knowledge) for MI455X / gfx1250.
// Memory/atomic-bound workload (~11 GFLOP vs multi-GB edge traffic, all
// L2-resident); matmuls done with exact-fp32 V_WMMA_F32_16X16X4_F32.
// Algebraic rewrite: segsum(jk[src]) @ Wout == segsum((jk @ Wout)[src]),
// so per-layer reps are immediately projected to 64 cols and only [E,64]
// is aggregated at the end (14x less traffic than materializing jk).

#define NN 30000
#define NE 300000
#define D_IN 256
#define D_H 128
#define D_OUT 64
#define N_LAYERS 6

typedef __attribute__((ext_vector_type(2))) float v2f;
typedef __attribute__((ext_vector_type(8))) float v8f;

// WMMA wrapper: the builtin only exists in the device compilation pass.
// Host pass gets a dummy body (never executed) so sema succeeds.
static __device__ inline v8f wmma_f32_16x16x4(v2f a, v2f b, v8f c) {
#if defined(__HIP_DEVICE_COMPILE__)
#if __has_builtin(__builtin_amdgcn_wmma_f32_16x16x4_f32)
  return __builtin_amdgcn_wmma_f32_16x16x4_f32(false, a, false, b, (short)0, c,
                                               false, false);
#else
#error "device pass: __builtin_amdgcn_wmma_f32_16x16x4_f32 missing - switch to bf16 16x16x32"
#endif
#else
  (void)a; (void)b;
  return c;
#endif
}

__global__ void k_fill(float* __restrict__ p, float v, int n) {
  int i = blockIdx.x * blockDim.x + threadIdx.x;
  if (i < n) p[i] = v;
}

__global__ void k_deg(const int* __restrict__ src, const int* __restrict__ dst,
                      float* __restrict__ odeg, float* __restrict__ ideg) {
  int i = blockIdx.x * blockDim.x + threadIdx.x;
  if (i < NE) {
    atomicAdd(&odeg[src[i]], 1.0f);
    atomicAdd(&ideg[dst[i]], 1.0f);
  }
}

__global__ void k_rsqrt(float* __restrict__ d, int n) {
  int i = blockIdx.x * blockDim.x + threadIdx.x;
  if (i < n) d[i] = rsqrtf(fmaxf(d[i], 1.0f));
}

// m[dst] += x[src] * inv_sqrt_out[src]   (one thread per (edge, float4-group))
template <int D>
__global__ void k_scatter(const float* __restrict__ x, const int* __restrict__ src,
                          const int* __restrict__ dst, const float* __restrict__ inv_out,
                          float* __restrict__ m) {
  constexpr int PER = D / 4;
  int g = blockIdx.x * blockDim.x + threadIdx.x;
  int e = g / PER;
  int f = (g % PER) * 4;
  if (e >= NE) return;
  int s = src[e], t = dst[e];
  float sc = inv_out[s];
  const float4 v = *(const float4*)(x + (size_t)s * D + f);
  float* mp = m + (size_t)t * D + f;
  atomicAdd(mp + 0, v.x * sc);
  atomicAdd(mp + 1, v.y * sc);
  atomicAdd(mp + 2, v.z * sc);
  atomicAdd(mp + 3, v.w * sc);
}

// H = relu((A * inv_in[:,None]) @ W + bias), A:[NN,K], W:[K,128], H:[NN,128]
// block = 256 threads = 8 waves; each wave owns one 16x16 output tile.
// grid.x = NN/16 (exact), so EXEC is all-ones for every WMMA.
__global__ void __launch_bounds__(256) k_gemm_relu(
    const float* __restrict__ A, const float* __restrict__ inv_in,
    const float* __restrict__ W, const float* __restrict__ bias,
    float* __restrict__ H, int K) {
  const int lane = threadIdx.x & 31;
  const int wave = threadIdx.x >> 5;   // 0..7 -> col tile
  const int half = lane >> 4;          // 0/1: K-subgroup per ISA A/B layout
  const int l16  = lane & 15;
  const int row0 = blockIdx.x * 16;
  const int col0 = wave * 16;

  const int arow = row0 + l16;
  const float asc = inv_in[arow];
  const float* ap = A + (size_t)arow * K + half * 2;           // A[row][k0+2*half ..]
  const float* bp = W + (size_t)(half * 2) * D_H + col0 + l16; // W[k0+2*half][col]

  v8f c = {};
  for (int k0 = 0; k0 < K; k0 += 4) {
    v2f a; a.x = ap[0] * asc; a.y = ap[1] * asc;
    v2f b; b.x = bp[0];       b.y = bp[D_H];
    c = wmma_f32_16x16x4(a, b, c);
    ap += 4;
    bp += 4 * D_H;
  }
#pragma unroll
  for (int r = 0; r < 8; ++r) {
    int row = row0 + r + half * 8;
    float v = c[r] + bias[col0 + l16];
    H[(size_t)row * D_H + col0 + l16] = fmaxf(v, 0.0f);
  }
}

// P += Hm @ Wo, Hm:[NN,128], Wo:[128,64], P:[NN,64]
// block = 128 threads = 4 waves (4 col tiles of 16); grid.x = NN/16.
__global__ void __launch_bounds__(128) k_gemm_acc(
    const float* __restrict__ Hm, const float* __restrict__ Wo,
    float* __restrict__ P) {
  const int lane = threadIdx.x & 31;
  const int wave = threadIdx.x >> 5;   // 0..3
  const int half = lane >> 4;
  const int l16  = lane & 15;
  const int row0 = blockIdx.x * 16;
  const int col  = wave * 16 + l16;

  const float* ap = Hm + (size_t)(row0 + l16) * D_H + half * 2;
  const float* bp = Wo + (size_t)(half * 2) * D_OUT + col;

  v8f c;
#pragma unroll
  for (int r = 0; r < 8; ++r)
    c[r] = P[(size_t)(row0 + r + half * 8) * D_OUT + col];

  for (int k0 = 0; k0 < D_H; k0 += 4) {
    v2f a; a.x = ap[0]; a.y = ap[1];
    v2f b; b.x = bp[0]; b.y = bp[D_OUT];
    c = wmma_f32_16x16x4(a, b, c);
    ap += 4;
    bp += 4 * D_OUT;
  }
#pragma unroll
  for (int r = 0; r < 8; ++r)
    P[(size_t)(row0 + r + half * 8) * D_OUT + col] = c[r];
}

__global__ void k_out_init(const float* __restrict__ bout, float* __restrict__ out) {
  int i = blockIdx.x * blockDim.x + threadIdx.x;
  if (i < NN * D_OUT) out[i] = bout[i & (D_OUT - 1)];
}

// out[dst] += P[src]  over 64 features
__global__ void k_scatter_out(const float* __restrict__ P, const int* __restrict__ src,
                              const int* __restrict__ dst, float* __restrict__ out) {
  constexpr int PER = D_OUT / 4;  // 16
  int g = blockIdx.x * blockDim.x + threadIdx.x;
  int e = g / PER;
  int f = (g % PER) * 4;
  if (e >= NE) return;
  int s = src[e], t = dst[e];
  const float4 v = *(const float4*)(P + (size_t)s * D_OUT + f);
  float* op = out + (size_t)t * D_OUT + f;
  atomicAdd(op + 0, v.x);
  atomicAdd(op + 1, v.y);
  atomicAdd(op + 2, v.z);
  atomicAdd(op + 3, v.w);
}

static inline int nblk(long n, int b) { return (int)((n + b - 1) / b); }

extern "C" void kernel_launch(void* const* d_in, const int* in_sizes, int n_in,
                              void* d_out, int out_size, void* d_ws, size_t ws_size,
                              hipStream_t stream) {
  (void)in_sizes; (void)n_in; (void)out_size; (void)ws_size;
  const float* feats = (const float*)d_in[0];
  const int*   src   = (const int*)d_in[1];
  const int*   dst   = (const int*)d_in[2];
  const float* W0    = (const float*)d_in[3];
  const float* b0    = (const float*)d_in[4];
  const float* Wh    = (const float*)d_in[5];
  const float* bh    = (const float*)d_in[6];
  const float* Wout  = (const float*)d_in[7];
  const float* bout  = (const float*)d_in[8];
  float* out = (float*)d_out;

  // workspace layout (floats): total NN*(2 + 256 + 128 + 128 + 64) ~= 69.4 MB
  float* ws      = (float*)d_ws;
  float* inv_out = ws;                        // NN
  float* inv_in  = inv_out + NN;              // NN
  float* m       = inv_in + NN;               // NN * D_IN (max message width)
  float* h0      = m + (size_t)NN * D_IN;     // NN * D_H
  float* h1      = h0 + (size_t)NN * D_H;     // NN * D_H
  float* p       = h1 + (size_t)NN * D_H;     // NN * D_OUT

  // degrees -> inv-sqrt (in place), zero projection accumulator
  k_fill<<<nblk(2L * NN, 256), 256, 0, stream>>>(inv_out, 0.0f, 2 * NN);
  k_deg<<<nblk(NE, 256), 256, 0, stream>>>(src, dst, inv_out, inv_in);
  k_rsqrt<<<nblk(2L * NN, 256), 256, 0, stream>>>(inv_out, 2 * NN);
  k_fill<<<nblk((long)NN * D_OUT, 256), 256, 0, stream>>>(p, 0.0f, NN * D_OUT);

  // layer 0: feats [N,256] -> h0 [N,128], then project into p
  k_fill<<<nblk((long)NN * D_IN, 256), 256, 0, stream>>>(m, 0.0f, NN * D_IN);
  k_scatter<D_IN><<<nblk((long)NE * (D_IN / 4), 256), 256, 0, stream>>>(
      feats, src, dst, inv_out, m);
  k_gemm_relu<<<NN / 16, 256, 0, stream>>>(m, inv_in, W0, b0, h0, D_IN);
  k_gemm_acc<<<NN / 16, 128, 0, stream>>>(h0, Wout, p);

  float* ha = h0;  // current rep
  float* hb = h1;
  for (int i = 0; i < N_LAYERS; ++i) {
    k_fill<<<nblk((long)NN * D_H, 256), 256, 0, stream>>>(m, 0.0f, NN * D_H);
    k_scatter<D_H><<<nblk((long)NE * (D_H / 4), 256), 256, 0, stream>>>(
        ha, src, dst, inv_out, m);
    k_gemm_relu<<<NN / 16, 256, 0, stream>>>(
        m, inv_in, Wh + (size_t)i * D_H * D_H, bh + (size_t)i * D_H, hb, D_H);
    k_gemm_acc<<<NN / 16, 128, 0, stream>>>(
        hb, Wout + (size_t)(i + 1) * D_H * D_OUT, p);
    float* t = ha; ha = hb; hb = t;
  }

  // out = bout + segment_sum(p[src], dst)
  k_out_init<<<nblk((long)NN * D_OUT, 256), 256, 0, stream>>>(bout, out);
  k_scatter_out<<<nblk((long)NE * (D_OUT / 4), 256), 256, 0, stream>>>(
      p, src, dst, out);
}